// Pointer_47347719471718
// MI455X (gfx1250) — compile-verified
//
#include <hip/hip_runtime.h>
#include <cmath>

#define B_ 128
#define S_ 2048
#define E_ 512
#define H_ 512

typedef __attribute__((ext_vector_type(2))) float v2f;
typedef __attribute__((ext_vector_type(8))) float v8f;

// ---------------------------------------------------------------------------
// WMMA f32 GEMM: C[M,N] = A[M,K] @ B (+ bias). One wave per 16x16 tile.
//   BT=true : Bm is [N,K] (i.e. we multiply by Bm^T)  -> used for query @ Wq^T
//   BT=false: Bm is [K,N]                             -> used for q @ Wk
// A 16x4 f32 layout: lane half selects K in {0,1} vs {2,3}; vgpr idx orders K.
// B 4x16 f32 layout: N striped across lanes, same K split across lane halves.
// C/D 16x16 f32: vgpr r -> M=r (lanes 0-15) / M=r+8 (lanes 16-31), N = lane&15.
// ---------------------------------------------------------------------------
template <bool BT, bool BIAS>
__global__ __launch_bounds__(32)
void wmma_gemm_f32(const float* __restrict__ A, const float* __restrict__ Bm,
                   const float* __restrict__ bias, float* __restrict__ C,
                   int M, int N, int K) {
  const int lane = threadIdx.x;            // 0..31, EXEC all ones
  const int half = lane >> 4;              // 0 or 1
  const int id   = lane & 15;
  const int tiles_n = N >> 4;
  const int tm = (blockIdx.x / tiles_n) << 4;
  const int tn = (blockIdx.x % tiles_n) << 4;

  const float* arow = A + (size_t)(tm + id) * K;
  const float* brow = BT ? (Bm + (size_t)(tn + id) * K) : Bm;

  v8f acc = {};
  for (int k0 = 0; k0 < K; k0 += 4) {
    const int ka = k0 + 2 * half;
    v2f a = *(const v2f*)(arow + ka);      // A[id][ka], A[id][ka+1]
    v2f b;
    if (BT) {
      b = *(const v2f*)(brow + ka);        // Bm[tn+id][ka], [ka+1]
    } else {
      const float* bp = Bm + (size_t)ka * N + (tn + id);
      b.x = bp[0];                         // Bm[ka][tn+id]
      b.y = bp[N];                         // Bm[ka+1][tn+id]
    }
    acc = __builtin_amdgcn_wmma_f32_16x16x4_f32(
        /*neg_a=*/false, a, /*neg_b=*/false, b,
        /*c_mod=*/(short)0, acc, /*reuse_a=*/false, /*reuse_b=*/false);
  }

  const float bv = BIAS ? bias[tn + id] : 0.0f;
#pragma unroll
  for (int r = 0; r < 8; ++r) {
    const int m = tm + r + 8 * half;
    C[(size_t)m * N + tn + id] = acc[r] + bv;
  }
}

// ---------------------------------------------------------------------------
// c[b] = dot(q[b,:], bk)   (one wave per b; trivial)
// ---------------------------------------------------------------------------
__global__ __launch_bounds__(32)
void dot_bias_kernel(const float* __restrict__ q, const float* __restrict__ bk,
                     float* __restrict__ c) {
  const int b = blockIdx.x;
  const int lane = threadIdx.x;
  float acc = 0.0f;
  for (int h = lane; h < H_; h += 32) acc = fmaf(q[(size_t)b * H_ + h], bk[h], acc);
  for (int off = 16; off > 0; off >>= 1) acc += __shfl_down(acc, off, 32);
  if (lane == 0) c[b] = acc;
}

// ---------------------------------------------------------------------------
// Streaming, bandwidth-bound kernel (reads all 512 MB of target once):
//   scores[b,s] = mask ? -inf : 10*tanh( dot(v[b,:], target[b,s,:]) + c[b] )
// One wave per score; lane l holds v[b, 4l + 128r] (r=0..3) in registers.
// Each wave-step issues 4 x global_load_b128 covering a contiguous 2 KB row.
// ---------------------------------------------------------------------------
__global__ __launch_bounds__(256)
void score_kernel(const float* __restrict__ v, const float* __restrict__ target,
                  const int* __restrict__ mask, const float* __restrict__ c,
                  float* __restrict__ scores) {
  const int b    = blockIdx.y;
  const int s0   = blockIdx.x * 128;       // 16 blocks along x cover S
  const int lane = threadIdx.x & 31;
  const int wave = threadIdx.x >> 5;       // 8 waves, 16 rows each

  float4 vv[4];
  const float4* vB = (const float4*)(v + (size_t)b * E_);
#pragma unroll
  for (int r = 0; r < 4; ++r) vv[r] = vB[lane + 32 * r];
  const float cb = c[b];

  for (int i = 0; i < 16; ++i) {
    const int s = s0 + wave * 16 + i;
    const float4* t = (const float4*)(target + ((size_t)b * S_ + s) * (size_t)E_);
    if (s + 1 < S_) __builtin_prefetch(t + 128, 0, 1);  // next row (speculative)
    float acc = 0.0f;
#pragma unroll
    for (int r = 0; r < 4; ++r) {
      const float4 tv = t[lane + 32 * r];
      acc = fmaf(vv[r].x, tv.x, acc);
      acc = fmaf(vv[r].y, tv.y, acc);
      acc = fmaf(vv[r].z, tv.z, acc);
      acc = fmaf(vv[r].w, tv.w, acc);
    }
    for (int off = 16; off > 0; off >>= 1) acc += __shfl_down(acc, off, 32);
    if (lane == 0) {
      float qk = 10.0f * tanhf(acc + cb);
      if (mask[(size_t)b * S_ + s] == 1) qk = -__builtin_inff();
      scores[(size_t)b * S_ + s] = qk;
    }
  }
}

// ---------------------------------------------------------------------------
// Row softmax over S=2048 (one block per b; 2 MB total traffic, negligible)
// ---------------------------------------------------------------------------
__global__ __launch_bounds__(256)
void softmax_kernel(const float* __restrict__ scores, float* __restrict__ out) {
  const int b = blockIdx.x;
  const int tid = threadIdx.x;
  const int lane = tid & 31, w = tid >> 5;
  __shared__ float red[8];
  const float* row = scores + (size_t)b * S_;
  float* orow = out + (size_t)b * S_;

  float m = -__builtin_inff();
  for (int s = tid; s < S_; s += 256) m = fmaxf(m, row[s]);
  for (int off = 16; off > 0; off >>= 1) m = fmaxf(m, __shfl_down(m, off, 32));
  if (lane == 0) red[w] = m;
  __syncthreads();
  float bm = red[0];
#pragma unroll
  for (int i = 1; i < 8; ++i) bm = fmaxf(bm, red[i]);
  __syncthreads();

  float sum = 0.0f;
  for (int s = tid; s < S_; s += 256) {
    const float e = __expf(row[s] - bm);
    orow[s] = e;
    sum += e;
  }
  for (int off = 16; off > 0; off >>= 1) sum += __shfl_down(sum, off, 32);
  if (lane == 0) red[w] = sum;
  __syncthreads();
  float tot = 0.0f;
#pragma unroll
  for (int i = 0; i < 8; ++i) tot += red[i];
  const float inv = 1.0f / tot;
  for (int s = tid; s < S_; s += 256) orow[s] *= inv;
}

// ---------------------------------------------------------------------------
extern "C" void kernel_launch(void* const* d_in, const int* in_sizes, int n_in,
                              void* d_out, int out_size, void* d_ws, size_t ws_size,
                              hipStream_t stream) {
  (void)in_sizes; (void)n_in; (void)out_size; (void)ws_size;
  const float* query  = (const float*)d_in[0];   // [B,E]
  const float* target = (const float*)d_in[1];   // [B,S,E]
  const int*   mask   = (const int*)  d_in[2];   // [B,S]
  const float* Wq     = (const float*)d_in[3];   // [H,E]
  const float* bq     = (const float*)d_in[4];   // [H]
  const float* Wk     = (const float*)d_in[5];   // [H,E]
  const float* bk     = (const float*)d_in[6];   // [H]
  float* out = (float*)d_out;                    // [B,S]

  // Workspace layout (all offsets 16B-aligned): q, v, c, scores
  float* q_ws  = (float*)d_ws;                         // B*H
  float* v_ws  = q_ws + (size_t)B_ * H_;               // B*E
  float* c_ws  = v_ws + (size_t)B_ * E_;               // B (padded to 256)
  float* sc_ws = c_ws + 256;                           // B*S

  // 1) q = query @ Wq^T + bq          (WMMA, 256 tiles)
  wmma_gemm_f32<true, true><<<(B_ / 16) * (H_ / 16), 32, 0, stream>>>(
      query, Wq, bq, q_ws, B_, H_, E_);
  // 2) v = q @ Wk                      (WMMA, 256 tiles)
  wmma_gemm_f32<false, false><<<(B_ / 16) * (E_ / 16), 32, 0, stream>>>(
      q_ws, Wk, nullptr, v_ws, B_, E_, H_);
  // 3) c[b] = q[b] . bk
  dot_bias_kernel<<<B_, 32, 0, stream>>>(q_ws, bk, c_ws);
  // 4) scores = mask/tanh( v . target + c )   -- the 512 MB streaming pass
  score_kernel<<<dim3(S_ / 128, B_), 256, 0, stream>>>(v_ws, target, mask, c_ws, sc_ws);
  // 5) row softmax
  softmax_kernel<<<B_, 256, 0, stream>>>(sc_ws, out);
}